// QTLayer_62234076119637
// MI455X (gfx1250) — compile-verified
//
#include <hip/hip_runtime.h>
#include <stdint.h>

// ---------------------------------------------------------------------------
// TT Q-function evaluation: out[b] = G0[0,i0,:] @ G1[:,i1,:] @ ... @ G7[:,i7,0]
// B = 262144, N = 256, R = 8, cores f32.
//
// Strategy (MI455X / gfx1250):
//  * Issue TDM first: G1..G4 staged by the Tensor Data Mover with a
//    transposing 3D descriptor ([r][n][s] -> [n][r][s]) plus LDS padding
//    (4 dwords per 64-dword matrix => 68-dword matrix stride, 16 bank-start
//    positions) so per-lane gathers are contiguous 16B-aligned ds_load_b128.
//  * Overlap: while the TDM streams 256KB into LDS, all 32 waves cooperatively
//    stage G0 (straight) and G7 (transposed) into LDS; then s_wait_tensorcnt
//    + workgroup barrier.
//  * G5/G6 read directly from global (L2-resident, 64KB each).
//  * Per-lane unrolled f32 FMA chain does the 8x8 matvecs (WMMA does not
//    apply: operands are per-lane, not wave-uniform).
// ---------------------------------------------------------------------------

typedef unsigned int u32x4 __attribute__((ext_vector_type(4)));
typedef int          i32x4 __attribute__((ext_vector_type(4)));
typedef int          i32x8 __attribute__((ext_vector_type(8)));

#define N_MODE 256
#define R_RANK 8
#define CORE_ELEMS (R_RANK * N_MODE * R_RANK)          // 16384 floats / core
#define LDS_MAT_STRIDE 68                              // 64 data + 4 pad dwords
#define LDS_CORE_STRIDE (N_MODE * LDS_MAT_STRIDE)      // 17408 floats
#define LDS_G0_OFF (4 * LDS_CORE_STRIDE)               // 69632
#define LDS_G7_OFF (LDS_G0_OFF + N_MODE * R_RANK)      // 71680
#define LDS_TOTAL_FLOATS (LDS_G7_OFF + N_MODE * R_RANK) // 73728 (= 294912 B)

#if defined(__HIP_DEVICE_COMPILE__) && \
    __has_builtin(__builtin_amdgcn_tensor_load_to_lds) && \
    __has_builtin(__builtin_amdgcn_s_wait_tensorcnt)
#define HAVE_TDM 1
#else
#define HAVE_TDM 0
#endif

#if HAVE_TDM
// DMA one (R,N,R) f32 core from global into LDS, transposed to [n][r][s]
// with a 4-dword pad after each 64-dword matrix (68-dword matrix stride).
// Descriptor: X = s (8, contiguous), Y = r (8, stride 2048 elems),
//             Z = n (256, stride 8 elems); LDS written sequentially + pad.
__device__ __forceinline__ void tdm_load_core_transposed(const float* gsrc,
                                                         uint32_t lds_byte_addr) {
  const uint64_t ga = (uint64_t)(uintptr_t)gsrc;
  // Group 0: count=1 (valid), lds_addr, global_addr[56:0], type=2
  const u32x4 d0 = { 1u,
                     lds_byte_addr,
                     (uint32_t)ga,
                     (((uint32_t)(ga >> 32)) & 0x01FFFFFFu) | 0x80000000u };
  // Group 1:
  //  w0: data_size=2 (4B) | pad_enable | pad_interval=5 (64 dw) | pad_amount=3 (4 dw)
  //  w1: tensor_dim0[15:0]=8 in [31:16]
  //  w2: tensor_dim1[15:0]=8 in [31:16]
  //  w3: tile_dim0=8 in [31:16]
  //  w4: tile_dim1=8 | tile_dim2=256<<16
  //  w5: tensor_dim0_stride[31:0]=2048
  //  w6: tensor_dim1_stride[15:0]=8 in [31:16]
  const i32x8 d1 = { (int)0x07520000,
                     (int)(8u << 16),
                     (int)(8u << 16),
                     (int)(8u << 16),
                     (int)(8u | (256u << 16)),
                     2048,
                     (int)(8u << 16),
                     0 };
  // Group 2: tensor_dim2 = 256 (covers the Z loop), no iteration, tile_dim3=0
  const i32x4 d2 = { 256, 0, 0, 0 };
  const i32x4 d3 = { 0, 0, 0, 0 };
#if defined(__clang_major__) && (__clang_major__ >= 23)
  const i32x8 d4 = { 0, 0, 0, 0, 0, 0, 0, 0 };
  __builtin_amdgcn_tensor_load_to_lds(d0, d1, d2, d3, d4, 0);
#else
  __builtin_amdgcn_tensor_load_to_lds(d0, d1, d2, d3, 0);
#endif
}
#endif // HAVE_TDM

// o[s] = sum_r v[r] * m[r*rowStride + s], 8x8, fully unrolled.
// m may be an LDS or a global pointer (address space resolved after inlining).
__device__ __forceinline__ void matvec8(const float* __restrict__ m, int rowStride,
                                        const float* __restrict__ v,
                                        float* __restrict__ o) {
  float a0 = 0.f, a1 = 0.f, a2 = 0.f, a3 = 0.f;
  float a4 = 0.f, a5 = 0.f, a6 = 0.f, a7 = 0.f;
#pragma unroll
  for (int r = 0; r < 8; ++r) {
    const float4 m0 = *(const float4*)(m + r * rowStride);
    const float4 m1 = *(const float4*)(m + r * rowStride + 4);
    const float vr = v[r];
    a0 = fmaf(vr, m0.x, a0);
    a1 = fmaf(vr, m0.y, a1);
    a2 = fmaf(vr, m0.z, a2);
    a3 = fmaf(vr, m0.w, a3);
    a4 = fmaf(vr, m1.x, a4);
    a5 = fmaf(vr, m1.y, a5);
    a6 = fmaf(vr, m1.z, a6);
    a7 = fmaf(vr, m1.w, a7);
  }
  o[0] = a0; o[1] = a1; o[2] = a2; o[3] = a3;
  o[4] = a4; o[5] = a5; o[6] = a6; o[7] = a7;
}

__global__ __launch_bounds__(1024) void qtt_tdm_kernel(
    const float* __restrict__ G0, const float* __restrict__ G1,
    const float* __restrict__ G2, const float* __restrict__ G3,
    const float* __restrict__ G4, const float* __restrict__ G5,
    const float* __restrict__ G6, const float* __restrict__ G7,
    const int* __restrict__ states, const int* __restrict__ actions,
    float* __restrict__ out, int B) {
  extern __shared__ __align__(16) float smem[];
  const int tid = (int)threadIdx.x;

#if HAVE_TDM
  // ---- Issue the four TDM descriptors FIRST so the DMA overlaps with the
  // cooperative staging below. Wave 0 issues; TENSORcnt is per-wave.
  if (tid == 0) {
    tdm_load_core_transposed(G1, (uint32_t)(uintptr_t)(smem + 0 * LDS_CORE_STRIDE));
    tdm_load_core_transposed(G2, (uint32_t)(uintptr_t)(smem + 1 * LDS_CORE_STRIDE));
    tdm_load_core_transposed(G3, (uint32_t)(uintptr_t)(smem + 2 * LDS_CORE_STRIDE));
    tdm_load_core_transposed(G4, (uint32_t)(uintptr_t)(smem + 3 * LDS_CORE_STRIDE));
  }
#endif

  // ---- Stage G0 (already [n][r] contiguous) and G7 (transposed to [n][r])
  // cooperatively; runs concurrently with the TDM transfers.
  {
    const float4* s4 = (const float4*)G0;
    float4* d4p = (float4*)(smem + LDS_G0_OFF);
    if (tid < 512) d4p[tid] = s4[tid];           // 2048 floats
#pragma unroll 2
    for (int i = tid; i < N_MODE * R_RANK; i += 1024) {
      const int n = i >> 3, r = i & 7;
      smem[LDS_G7_OFF + i] = G7[r * N_MODE + n];
    }
  }

#if HAVE_TDM
  if (tid == 0) {
    __builtin_amdgcn_s_wait_tensorcnt(0);        // TDM data landed in LDS
  }
#else
  // ---- Fallback: cooperative transposed + padded staging of G1..G4.
  {
    const float* gsrc[4] = { G1, G2, G3, G4 };
    for (int t = tid; t < 4 * CORE_ELEMS; t += 1024) {
      const int k = t >> 14;                 // core
      const int e = t & (CORE_ELEMS - 1);
      const int n = e >> 6;                  // matrix index
      const int rs = e & 63;
      const int r = rs >> 3, s = rs & 7;
      smem[k * LDS_CORE_STRIDE + n * LDS_MAT_STRIDE + r * 8 + s] =
          gsrc[k][r * (N_MODE * R_RANK) + n * R_RANK + s];
    }
  }
#endif
  __syncthreads();

  // ---- Per-element TT contraction.
  const int b = (int)blockIdx.x * 1024 + tid;
  if (b < B) {
    const int* st = states + b * 7;
    const int i0 = st[0], i1 = st[1], i2 = st[2], i3 = st[3];
    const int i4 = st[4], i5 = st[5], i6 = st[6];
    const int i7 = actions[b];

    float v[8], t[8];
    {
      const float4 a0 = *(const float4*)(smem + LDS_G0_OFF + i0 * 8);
      const float4 a1 = *(const float4*)(smem + LDS_G0_OFF + i0 * 8 + 4);
      v[0] = a0.x; v[1] = a0.y; v[2] = a0.z; v[3] = a0.w;
      v[4] = a1.x; v[5] = a1.y; v[6] = a1.z; v[7] = a1.w;
    }
    // Cores 1..4 from LDS (transposed, 68-dword skewed layout).
    matvec8(smem + 0 * LDS_CORE_STRIDE + i1 * LDS_MAT_STRIDE, 8, v, t);
    matvec8(smem + 1 * LDS_CORE_STRIDE + i2 * LDS_MAT_STRIDE, 8, t, v);
    matvec8(smem + 2 * LDS_CORE_STRIDE + i3 * LDS_MAT_STRIDE, 8, v, t);
    matvec8(smem + 3 * LDS_CORE_STRIDE + i4 * LDS_MAT_STRIDE, 8, t, v);
    // Cores 5..6 from global (L2-resident), original [r][n][s] layout.
    matvec8(G5 + i5 * 8, N_MODE * R_RANK, v, t);
    matvec8(G6 + i6 * 8, N_MODE * R_RANK, t, v);
    // Final dot with G7 column (transposed in LDS).
    const float4 c0 = *(const float4*)(smem + LDS_G7_OFF + i7 * 8);
    const float4 c1 = *(const float4*)(smem + LDS_G7_OFF + i7 * 8 + 4);
    const float q = fmaf(v[0], c0.x,
                    fmaf(v[1], c0.y,
                    fmaf(v[2], c0.z,
                    fmaf(v[3], c0.w,
                    fmaf(v[4], c1.x,
                    fmaf(v[5], c1.y,
                    fmaf(v[6], c1.z, v[7] * c1.w)))))));
    out[b] = q;
  }
}

extern "C" void kernel_launch(void* const* d_in, const int* in_sizes, int n_in,
                              void* d_out, int out_size, void* d_ws, size_t ws_size,
                              hipStream_t stream) {
  (void)n_in; (void)d_ws; (void)ws_size;
  const float* G0 = (const float*)d_in[0];
  const float* G1 = (const float*)d_in[1];
  const float* G2 = (const float*)d_in[2];
  const float* G3 = (const float*)d_in[3];
  const float* G4 = (const float*)d_in[4];
  const float* G5 = (const float*)d_in[5];
  const float* G6 = (const float*)d_in[6];
  const float* G7 = (const float*)d_in[7];
  const int* states  = (const int*)d_in[8];
  const int* actions = (const int*)d_in[9];
  float* out = (float*)d_out;
  const int B = in_sizes[9];                 // actions: (B,)
  (void)out_size;

  const int block = 1024;
  const int grid = (B + block - 1) / block;  // 256 for B=262144
  const size_t shmem = (size_t)LDS_TOTAL_FLOATS * sizeof(float);  // 294912 B

  qtt_tdm_kernel<<<grid, block, shmem, stream>>>(G0, G1, G2, G3, G4, G5, G6, G7,
                                                 states, actions, out, B);
}